// Attn_59210419142961
// MI455X (gfx1250) — compile-verified
//
#include <hip/hip_runtime.h>
#include <hip/hip_bf16.h>

typedef __attribute__((ext_vector_type(2))) float v2f;
typedef __attribute__((ext_vector_type(8))) float v8f;

#define BB 8
#define LL 512
#define DD 128
#define MM (BB * LL)   // 4096 rows for the projection GEMM

__device__ __forceinline__ float dev_tanhf(float x) {
#if __has_builtin(__builtin_amdgcn_tanhf)
    return __builtin_amdgcn_tanhf(x);
#elif __has_builtin(__builtin_amdgcn_tanh_f32)
    return __builtin_amdgcn_tanh_f32(x);
#else
    // Guaranteed single TRANS op. v_nop satisfies the CDNA5 TRANS hazard:
    // 1 independent op before output use / source overwrite (ISA 7.4 case 3).
    float r;
    asm volatile("v_tanh_f32 %0, %1\n\tv_nop" : "=v"(r) : "v"(x));
    return r;
#endif
}

// -------------------------------------------------------------------------
// Kernel 2 (placed first so the disasm snippet shows its inner loop):
// fused score + tanh + softmax.
// One workgroup (256 threads) per (b, i) row:
//   score[j] = sum_d a[d] * tanh(q[b,j,d] + k[b,i,d]),  softmax over j.
// Thread t owns j = t and j = t+256.
// -------------------------------------------------------------------------
__global__ __launch_bounds__(256) void score_softmax_kernel(
    const float* __restrict__ Q,     // [B,L,D]  (u projection, indexed by j)
    const float* __restrict__ Kt,    // [B,L,D]  (v projection, indexed by i)
    const float* __restrict__ Aw,    // [D]
    float* __restrict__ AttnOut)     // [B,L,L]
{
    __shared__ float kv[DD];
    __shared__ float av[DD];
    __shared__ float red[8];

    int b = blockIdx.x >> 9;         // / LL
    int i = blockIdx.x & (LL - 1);
    int t = threadIdx.x;

    if (t < DD) {
        kv[t] = Kt[((size_t)b * LL + i) * DD + t];
        av[t] = Aw[t];
    }
    __syncthreads();

    float s[2];
#pragma unroll
    for (int jj = 0; jj < 2; ++jj) {
        int j = t + jj * 256;
        const float* q = Q + ((size_t)b * LL + j) * DD;
        float acc = 0.0f;
#pragma unroll 4
        for (int d = 0; d < DD; d += 4) {
            float4 qv = *(const float4*)(q + d);
            acc += av[d + 0] * dev_tanhf(qv.x + kv[d + 0]);
            acc += av[d + 1] * dev_tanhf(qv.y + kv[d + 1]);
            acc += av[d + 2] * dev_tanhf(qv.z + kv[d + 2]);
            acc += av[d + 3] * dev_tanhf(qv.w + kv[d + 3]);
        }
        s[jj] = acc;
    }

    // --- row max (wave32 shuffle + LDS across 8 waves) ---
    float m = fmaxf(s[0], s[1]);
#pragma unroll
    for (int off = 16; off > 0; off >>= 1)
        m = fmaxf(m, __shfl_xor(m, off, 32));
    int wid = t >> 5, lane = t & 31;
    if (lane == 0) red[wid] = m;
    __syncthreads();
    float M = red[0];
#pragma unroll
    for (int w = 1; w < 8; ++w) M = fmaxf(M, red[w]);

    // --- exp + row sum ---
    float e0 = __expf(s[0] - M);
    float e1 = __expf(s[1] - M);
    float sm = e0 + e1;
#pragma unroll
    for (int off = 16; off > 0; off >>= 1)
        sm += __shfl_xor(sm, off, 32);
    __syncthreads();                 // everyone done reading red[]
    if (lane == 0) red[wid] = sm;
    __syncthreads();
    float S = red[0];
#pragma unroll
    for (int w = 1; w < 8; ++w) S += red[w];
    float inv = 1.0f / S;

    float* arow = AttnOut + ((size_t)b * LL + i) * LL;
    arow[t]       = e0 * inv;
    arow[t + 256] = e1 * inv;
}

// -------------------------------------------------------------------------
// Kernel 1: projections.  Y[m,n] = sum_k X[m,k] * W[n,k]  (x @ W^T)
// M=4096, N=K=128 compile-time.  fp32 WMMA 16x16x4, one wave per 16x16 tile.
// gridDim.y selects (u_w -> q) vs (v_w -> k).
// A frag (16x4 f32): lanes 0-15 hold {K0,K1}, lanes 16-31 hold {K2,K3}.
// B frag (4x16 f32): mirrored; B = W^T so lanes load W rows identically.
// C/D (16x16 f32): VGPR v -> M = v (lanes 0-15) / v+8 (lanes 16-31), N = lane&15.
// -------------------------------------------------------------------------
__global__ __launch_bounds__(128) void proj_wmma_kernel(
    const float* __restrict__ X,
    const float* __restrict__ W0, const float* __restrict__ W1,
    float* __restrict__ Y0, float* __restrict__ Y1)
{
    const float* W = blockIdx.y ? W1 : W0;
    float*       Y = blockIdx.y ? Y1 : Y0;

    int wave = threadIdx.x >> 5;               // 4 waves / block
    int lane = threadIdx.x & 31;
    int tile = blockIdx.x * 4 + wave;
    const int tiles_n = DD >> 4;               // 8
    int m0 = (tile / tiles_n) << 4;
    int n0 = (tile % tiles_n) << 4;

    int koff = (lane >> 4) << 1;               // 0 or 2
    const float* pa = X + (size_t)(m0 + (lane & 15)) * DD + koff;
    const float* pb = W + (size_t)(n0 + (lane & 15)) * DD + koff;

    v8f c = {};
#pragma unroll 8
    for (int k = 0; k < DD; k += 4) {
        v2f a = *(const v2f*)(pa + k);
        v2f b = *(const v2f*)(pb + k);
        c = __builtin_amdgcn_wmma_f32_16x16x4_f32(false, a, false, b,
                                                  (short)0, c, false, false);
    }
    int mrow = m0 + ((lane >> 4) << 3);
    int col  = n0 + (lane & 15);
#pragma unroll
    for (int v = 0; v < 8; ++v)
        Y[(size_t)(mrow + v) * DD + col] = c[v];
}

// -------------------------------------------------------------------------
// Kernel 3: out[b,i,d] = sum_j attn[b,i,j] * inp[b,j,d]
// Per batch: M=512 (i), K=512 (j), N=128 (d). fp32 WMMA 16x16x4.
// A frag along attn rows (contiguous K), B frag along inp columns (K-strided).
// -------------------------------------------------------------------------
__global__ __launch_bounds__(128) void av_wmma_kernel(
    const float* __restrict__ Attn,   // [B,L,L]
    const float* __restrict__ Xin,    // [B,L,D]
    float* __restrict__ Out)          // [B,L,D]
{
    int wave = threadIdx.x >> 5;
    int lane = threadIdx.x & 31;
    int tile = blockIdx.x * 4 + wave;

    const int tiles_n  = DD >> 4;              // 8
    const int tiles_m  = LL >> 4;              // 32
    const int tiles_pb = tiles_m * tiles_n;    // 256
    int b  = tile / tiles_pb;
    int tt = tile % tiles_pb;
    int m0 = (tt / tiles_n) << 4;
    int n0 = (tt % tiles_n) << 4;

    const float* Ab = Attn + (size_t)b * LL * LL;
    const float* Xb = Xin  + (size_t)b * LL * DD;

    int koff = (lane >> 4) << 1;               // 0 or 2
    const float* pa = Ab + (size_t)(m0 + (lane & 15)) * LL + koff;
    const float* pb = Xb + (size_t)koff * DD + (n0 + (lane & 15));

    v8f c = {};
#pragma unroll 8
    for (int k = 0; k < LL; k += 4) {
        v2f a = *(const v2f*)(pa + k);
        v2f bf;
        bf.x = pb[(size_t)(k + 0) * DD];
        bf.y = pb[(size_t)(k + 1) * DD];
        c = __builtin_amdgcn_wmma_f32_16x16x4_f32(false, a, false, bf,
                                                  (short)0, c, false, false);
    }
    float* Ob = Out + (size_t)b * LL * DD;
    int mrow = m0 + ((lane >> 4) << 3);
    int col  = n0 + (lane & 15);
#pragma unroll
    for (int v = 0; v < 8; ++v)
        Ob[(size_t)(mrow + v) * DD + col] = c[v];
}

extern "C" void kernel_launch(void* const* d_in, const int* in_sizes, int n_in,
                              void* d_out, int out_size, void* d_ws, size_t ws_size,
                              hipStream_t stream)
{
    (void)in_sizes; (void)n_in; (void)out_size; (void)ws_size;
    const float* inp = (const float*)d_in[0];   // [B,L,D]
    const float* u_w = (const float*)d_in[1];   // [D,D]
    const float* v_w = (const float*)d_in[2];   // [D,D]
    const float* a_w = (const float*)d_in[3];   // [1,D]

    float* out  = (float*)d_out;                        // [B,L,D]
    float* attn = out + (size_t)BB * LL * DD;           // [B,L,L]

    float* q = (float*)d_ws;                            // [B,L,D]
    float* k = q + (size_t)BB * LL * DD;                // [B,L,D]

    const int proj_tiles = (MM / 16) * (DD / 16);       // 2048
    dim3 pgrid(proj_tiles / 4, 2);
    proj_wmma_kernel<<<pgrid, 128, 0, stream>>>(inp, u_w, v_w, q, k);

    score_softmax_kernel<<<BB * LL, 256, 0, stream>>>(q, k, a_w, attn);

    const int av_tiles = BB * (LL / 16) * (DD / 16);    // 2048
    av_wmma_kernel<<<av_tiles / 4, 128, 0, stream>>>(attn, inp, out);
}